// CNF_81226421502247
// MI455X (gfx1250) — compile-verified
//
#include <hip/hip_runtime.h>
#include <hip/hip_bf16.h>

typedef float v2f __attribute__((ext_vector_type(2)));
typedef float v8f __attribute__((ext_vector_type(8)));

#define IO_DIM 64
#define WIDTH  16
#define HID    128
#define BS     (IO_DIM * WIDTH)   // 1024
#define OUTDIM (2 * BS + WIDTH)   // 2064

__device__ __forceinline__ v8f wmma4(v2f a, v2f b, v8f c) {
    // D = A(16x4, f32) x B(4x16, f32) + C(16x16, f32)
    return __builtin_amdgcn_wmma_f32_16x16x4_f32(
        /*neg_a=*/false, a, /*neg_b=*/false, b,
        /*c_mod=*/(short)0, c, /*reuse_a=*/false, /*reuse_b=*/false);
}

// ---------------------------------------------------------------------------
// Phase 1: tiny hypernetwork.  ws[0..1023]=u, ws[1024..2047]=w,
// ws[2048..2063]=bb, ws[2064..2079]=uw.
// ---------------------------------------------------------------------------
__global__ __launch_bounds__(256) void cnf_hyper(
    const float* __restrict__ t,  const float* __restrict__ W1,
    const float* __restrict__ b1, const float* __restrict__ W2,
    const float* __restrict__ b2, const float* __restrict__ W3,
    const float* __restrict__ b3, float* __restrict__ ws) {
    __shared__ float h1[HID];
    __shared__ float h2[HID];
    __shared__ float uwb[OUTDIM];
    const int tid = threadIdx.x;
    const float tt = t[0];

    if (tid < HID) h1[tid] = tanhf(W1[tid] * tt + b1[tid]);
    __syncthreads();
    if (tid < HID) {
        float acc = b2[tid];
        #pragma unroll 8
        for (int j = 0; j < HID; ++j) acc += W2[tid * HID + j] * h1[j];
        h2[tid] = tanhf(acc);
    }
    __syncthreads();
    for (int o = tid; o < OUTDIM; o += 256) {
        float acc = b3[o];
        #pragma unroll 8
        for (int j = 0; j < HID; ++j) acc += W3[o * HID + j] * h2[j];
        uwb[o] = acc;
        ws[o]  = acc;
    }
    __syncthreads();
    if (tid < WIDTH) {
        float s = 0.f;
        #pragma unroll 8
        for (int k = 0; k < IO_DIM; ++k)
            s += uwb[tid * IO_DIM + k] * uwb[BS + tid * IO_DIM + k];
        ws[OUTDIM + tid] = s;
    }
}

// ---------------------------------------------------------------------------
// Phase 2: streaming WMMA over z.  One 16-row tile per wave iteration.
// ---------------------------------------------------------------------------
__global__ __launch_bounds__(256) void cnf_main(
    const float* __restrict__ z, const float* __restrict__ ws,
    float* __restrict__ out, int nrows) {
    __shared__ __align__(16) float lds[8][256];   // per-wave 16x16 th slab

    const int lane = threadIdx.x & 31;
    const int wave = threadIdx.x >> 5;
    const int n16  = lane & 15;
    const int hi   = lane >> 4;       // 0: lanes 0-15 (K=0,1), 1: lanes 16-31 (K=2,3)
    const int koff = hi * 2;

    const float* u  = ws;             // 16x64 row-major
    const float* w  = ws + BS;        // 16x64 row-major
    const float* bb = ws + 2 * BS;    // 16
    const float* uw = ws + OUTDIM;    // 16

    // B operand for GEMM1 (a = z @ w^T): B[k][n] = w[n][k], chunk kk covers k=4kk..4kk+3.
    v2f wB[16];
    #pragma unroll
    for (int kk = 0; kk < 16; ++kk)
        wB[kk] = *(const v2f*)(w + n16 * 64 + 4 * kk + koff);

    // B operand for GEMM2 (dz = th @ u): group g covers cols 16g..16g+15.
    v2f uB[4][4];
    #pragma unroll
    for (int g = 0; g < 4; ++g)
        #pragma unroll
        for (int kk = 0; kk < 4; ++kk) {
            const int krow = 4 * kk + koff;
            v2f bv;
            bv.x = u[krow * 64 + 16 * g + n16];
            bv.y = u[(krow + 1) * 64 + 16 * g + n16];
            uB[g][kk] = bv;
        }

    const float bbv = bb[n16];
    const float uwv = uw[n16];

    float* out_dz = out;
    float* out_dl = out + (size_t)nrows * IO_DIM;

    const int ntiles = nrows >> 4;             // nrows multiple of 16 here
    const int gwave  = blockIdx.x * 8 + wave;  // wave-uniform -> EXEC all-1s
    const int nwaves = gridDim.x * 8;
    float* slab = lds[wave];

    for (int tile = gwave; tile < ntiles; tile += nwaves) {
        const int row_base = tile * 16;

        // ---- A operand (z tile, 16x64) in WMMA A-layout -------------------
        const float* zrow = z + (size_t)(row_base + n16) * 64 + koff;
        v2f a[16];
        #pragma unroll
        for (int kk = 0; kk < 16; ++kk)
            a[kk] = *(const v2f*)(zrow + 4 * kk);

        // ---- GEMM1: acc = z_tile @ w^T ------------------------------------
        v8f acc = {};
        #pragma unroll
        for (int kk = 0; kk < 16; ++kk)
            acc = wmma4(a[kk], wB[kk], acc);

        // ---- bias + tanh ---------------------------------------------------
        v8f th;
        #pragma unroll
        for (int v = 0; v < 8; ++v)
            th[v] = tanhf(acc[v] + bbv);

        // ---- trace: (1 - th^2) @ uw / 16, reduced across 16-lane halves ---
        float trv[8];
        #pragma unroll
        for (int v = 0; v < 8; ++v) {
            float p = (1.0f - th[v] * th[v]) * uwv;
            p += __shfl_xor(p, 1, 32);
            p += __shfl_xor(p, 2, 32);
            p += __shfl_xor(p, 4, 32);
            p += __shfl_xor(p, 8, 32);
            trv[v] = p;
        }
        if (n16 == 0) {
            #pragma unroll
            for (int v = 0; v < 8; ++v)
                out_dl[row_base + v + 8 * hi] = -trv[v] * 0.0625f;
        }

        // ---- transpose th from C/D layout to A layout via LDS -------------
        #pragma unroll
        for (int v = 0; v < 8; ++v)
            slab[(v + 8 * hi) * 16 + n16] = th[v];
        asm volatile("s_wait_dscnt 0" ::: "memory");  // cross-lane LDS RAW

        v2f a2[4];
        #pragma unroll
        for (int kk = 0; kk < 4; ++kk)
            a2[kk] = *(const v2f*)(slab + n16 * 16 + 4 * kk + koff);

        // ---- GEMM2: dz = th @ u / 16, 4 column groups ---------------------
        #pragma unroll
        for (int g = 0; g < 4; ++g) {
            v8f d = {};
            #pragma unroll
            for (int kk = 0; kk < 4; ++kk)
                d = wmma4(a2[kk], uB[g][kk], d);
            #pragma unroll
            for (int v = 0; v < 8; ++v)
                out_dz[(size_t)(row_base + v + 8 * hi) * 64 + 16 * g + n16] =
                    d[v] * 0.0625f;
        }
    }
}

// ---------------------------------------------------------------------------
extern "C" void kernel_launch(void* const* d_in, const int* in_sizes, int n_in,
                              void* d_out, int out_size, void* d_ws, size_t ws_size,
                              hipStream_t stream) {
    const float* t  = (const float*)d_in[0];
    const float* z  = (const float*)d_in[1];
    const float* W1 = (const float*)d_in[2];
    const float* b1 = (const float*)d_in[3];
    const float* W2 = (const float*)d_in[4];
    const float* b2 = (const float*)d_in[5];
    const float* W3 = (const float*)d_in[6];
    const float* b3 = (const float*)d_in[7];
    float* ws  = (float*)d_ws;
    float* out = (float*)d_out;
    const int nrows = in_sizes[1] / IO_DIM;   // N

    cnf_hyper<<<1, 256, 0, stream>>>(t, W1, b1, W2, b2, W3, b3, ws);
    cnf_main<<<1024, 256, 0, stream>>>(z, ws, out, nrows);
}